// PCF_67551245631497
// MI455X (gfx1250) — compile-verified
//
#include <hip/hip_runtime.h>
#include <math.h>

#define NBBINS 50
#define JCH 8            // j-chunks -> deterministic partial histograms (no fp atomics)

typedef float v2f __attribute__((ext_vector_type(2)));
typedef float v8f __attribute__((ext_vector_type(8)));

// ---------------------------------------------------------------------------
// Kernel A: normalized pairwise distance matrix via V_WMMA_F32_16X16X4_F32.
// Coords are centered at 0.5 (translation-invariant) then scaled by 1/RMAX to
// minimize cancellation in |a|^2 + |b|^2 - 2 a.b.
//   row_j (A, M-dim = j) = [xj, yj, xj^2+yj^2, 1]
//   col_i (B, N-dim = i) = [-2xi, -2yi, 1, xi^2+yi^2]
// Output stored transposed DmatT[j][i] = sqrt(D2) so kernel B reads coalesced.
// A-layout (32b 16x4): lanes 0-15 -> K=0,1 in v[0],v[1]; lanes 16-31 -> K=2,3.
// B-layout (32b 4x16): lanes 0-15 -> rows K=0,1; lanes 16-31 -> rows K=2,3.
// C/D layout: reg r, lane L -> M = r + 8*(L>>4), N = L&15.
// ---------------------------------------------------------------------------
__global__ void pcf_dist_wmma(const float* __restrict__ dA,
                              const float* __restrict__ dB,
                              const int* __restrict__ sameFlag,
                              float* __restrict__ DmatT,
                              int Na, int Nb, float invR, int jTmax)
{
    const int same = *sameFlag;
    const float* __restrict__ src = same ? dA : dB;   // rows = "j" points
    const int NbE = same ? Na : Nb;
    const int iT  = (Na + 15) >> 4;
    const int jT  = (NbE + 15) >> 4;

    const int wave = (blockIdx.x * blockDim.x + threadIdx.x) >> 5;
    const int lane = threadIdx.x & 31;
    const int jt = wave % jTmax;
    const int it = wave / jTmax;
    if (it >= iT || jt >= jT) return;          // wave-uniform: EXEC stays all-1s

    const int half = lane >> 4;                // 0: K=0,1  1: K=2,3
    const int m    = lane & 15;

    // A fragment: rows are j-points
    int jr = jt * 16 + m; jr = jr < NbE ? jr : (NbE - 1);
    float xj = (src[jr * 3 + 0] - 0.5f) * invR;
    float yj = (src[jr * 3 + 1] - 0.5f) * invR;
    v2f afrag;
    if (half == 0) { afrag[0] = xj;              afrag[1] = yj;   }
    else           { afrag[0] = xj*xj + yj*yj;   afrag[1] = 1.0f; }

    // B fragment: cols are i-points
    int ic = it * 16 + m; ic = ic < Na ? ic : (Na - 1);
    float xi = (dA[ic * 3 + 0] - 0.5f) * invR;
    float yi = (dA[ic * 3 + 1] - 0.5f) * invR;
    v2f bfrag;
    if (half == 0) { bfrag[0] = -2.0f * xi;      bfrag[1] = -2.0f * yi;    }
    else           { bfrag[0] = 1.0f;            bfrag[1] = xi*xi + yi*yi; }

    v8f c = {};
    // 8 args: (neg_a, A, neg_b, B, c_mod, C, reuse_a, reuse_b)
    c = __builtin_amdgcn_wmma_f32_16x16x4_f32(false, afrag, false, bfrag,
                                              (short)0, c, false, false);

    const bool fullTile = ((jt * 16 + 16) <= NbE) && ((it * 16 + 16) <= Na); // uniform
    const size_t base = (size_t)(jt * 16 + (half << 3)) * Na + (it * 16 + m);
    if (fullTile) {
        #pragma unroll
        for (int r = 0; r < 8; ++r)
            DmatT[base + (size_t)r * Na] = __builtin_amdgcn_sqrtf(fmaxf(c[r], 0.0f));
    } else {
        #pragma unroll
        for (int r = 0; r < 8; ++r) {
            int j = jt * 16 + r + (half << 3);
            int i = it * 16 + m;
            if (j < NbE && i < Na)
                DmatT[(size_t)j * Na + i] = __builtin_amdgcn_sqrtf(fmaxf(c[r], 0.0f));
        }
    }
}

// ---------------------------------------------------------------------------
// Kernel B: per-point soft histogram. lane <-> i, hist[50] in VGPRs,
// stream DmatT rows (coalesced), disks_b radius is wave-uniform (SMEM path).
// Writes partial hist part[jc][k][i] (k-major -> coalesced).
// ---------------------------------------------------------------------------
__global__ void pcf_hist(const float* __restrict__ dA,
                         const float* __restrict__ dB,
                         const int* __restrict__ sameFlag,
                         const float* __restrict__ DmatT,
                         float* __restrict__ part,
                         int Na, int Nb, float invR)
{
    const int same = *sameFlag;
    const int NbE  = same ? Na : Nb;
    const float* __restrict__ srcB = same ? dA : dB;

    const int jc = blockIdx.x % JCH;
    const int ib = blockIdx.x / JCH;
    const int i  = (ib << 8) + threadIdx.x;
    const bool valid = (i < Na);
    const int ii = valid ? i : 0;

    const float ra = dA[ii * 3 + 2] * invR;

    const int chunk = (NbE + JCH - 1) / JCH;
    const int j0 = jc * chunk;
    const int j1 = min(j0 + chunk, NbE);

    float h[NBBINS];
    #pragma unroll
    for (int k = 0; k < NBBINS; ++k) h[k] = 0.0f;

    for (int j = j0; j < j1; ++j) {
        float d = DmatT[(size_t)j * Na + ii];
        __builtin_prefetch(&DmatT[(size_t)(j + 4) * Na + ii], 0, 3);
        float rb = srcB[j * 3 + 2] * invR;

        float r1 = fmaxf(ra, rb);
        float r2 = fminf(ra, rb);
        float extent  = fmaxf(d + r1 + r2, 2.0f * r1);
        float overlap = fminf(fmaxf(r1 + r2 - d, 0.0f), 2.0f * r2);
        float f  = extent - overlap + d + r1 - r2;
        float c1 = f / (4.0f * r1 - 4.0f * r2 + 1e-8f);
        float c2 = (f - 4.0f * r1 + 7.0f * r2) / (3.0f * r2 + 1e-8f);
        float c3 = f - 4.0f * r1 - 2.0f * r2 + 3.0f;
        float dd = (d <= r1 - r2) ? c1 : ((d <= r1 + r2) ? c2 : c3);

        float msk = (valid && !(same && j == i)) ? 1.0f : 0.0f;

        // val_k = exp(-(rs_k/RMAX - dd)^2 / sigma^2), sigma=0.25 -> 1/s^2 = 16
        #pragma unroll
        for (int k = 0; k < NBBINS; ++k) {
            float x = 0.1f * (float)(k + 1) - dd;
            h[k] = fmaf(msk, __expf(-16.0f * x * x), h[k]);
        }
    }

    if (valid) {
        #pragma unroll
        for (int k = 0; k < NBBINS; ++k)
            part[((size_t)jc * NBBINS + k) * Na + i] = h[k];
    }
}

// ---------------------------------------------------------------------------
// Kernel C: perimeter weights + density + column mean/min/max. One block per bin.
// ---------------------------------------------------------------------------
__device__ __forceinline__ float border_sub(float dx, float dy, float rk)
{
    if (rk > dx) {
        float ratio = dx / rk;
        float alpha = acosf(fminf(fmaxf(ratio, -1.0f), 1.0f));
        return fminf(alpha, atan2f(dy, dx)) + fminf(alpha, atan2f(1.0f - dy, dx));
    }
    return 0.0f;
}

__global__ void pcf_finish(const float* __restrict__ dA,
                           const float* __restrict__ part,
                           float* __restrict__ out,
                           int Na, int Nb, float rmax, float gf)
{
    const int k   = blockIdx.x;
    const int tid = threadIdx.x;

    const float TWO_PI = 6.28318530717958647692f;
    const float rsn   = 0.1f * (float)(k + 1);   // rs/RMAX
    const float rk    = rsn * rmax;              // actual radius
    const float inner = fmaxf(0.0f, rk - 0.5f * rmax);
    const float outer = rk + 0.5f * rmax;
    const float area  = 3.14159265358979323846f * (outer * outer - inner * inner);
    const float scale = gf / (area * (float)Nb);

    float sum = 0.0f, mn = 3.4e38f, mx = -3.4e38f;

    for (int i = tid; i < Na; i += blockDim.x) {
        float hist = 0.0f;
        #pragma unroll
        for (int c = 0; c < JCH; ++c)
            hist += part[((size_t)c * NBBINS + k) * Na + i];

        float x = dA[i * 3 + 0];
        float y = dA[i * 3 + 1];
        float full = TWO_PI;
        full -= border_sub(x,        y, rk);
        full -= border_sub(1.0f - x, y, rk);
        full -= border_sub(y,        x, rk);
        full -= border_sub(1.0f - y, x, rk);
        float perim = fminf(fmaxf(full / TWO_PI, 0.0f), 1.0f);
        float w = (perim > 1e-4f) ? (1.0f / perim) : 0.0f;

        float dens = w * hist * scale;
        sum += dens;
        mn = fminf(mn, dens);
        mx = fmaxf(mx, dens);
    }

    __shared__ float sS[256], sN[256], sX[256];
    sS[tid] = sum; sN[tid] = mn; sX[tid] = mx;
    __syncthreads();
    for (int s = 128; s > 0; s >>= 1) {
        if (tid < s) {
            sS[tid] += sS[tid + s];
            sN[tid] = fminf(sN[tid], sN[tid + s]);
            sX[tid] = fmaxf(sX[tid], sX[tid + s]);
        }
        __syncthreads();
    }
    if (tid == 0) {
        out[2 * k + 0] = rsn;
        out[2 * k + 1] = sS[0] / (float)Na;
        out[2 * NBBINS + k] = fminf(sN[0], 1e4f);
        out[3 * NBBINS + k] = fmaxf(sX[0], 0.0f);
    }
}

// ---------------------------------------------------------------------------
extern "C" void kernel_launch(void* const* d_in, const int* in_sizes, int n_in,
                              void* d_out, int out_size, void* d_ws, size_t ws_size,
                              hipStream_t stream)
{
    const float* dA   = (const float*)d_in[0];
    const float* dB   = (const float*)d_in[1];
    const int*   same = (const int*)d_in[2];

    const int Na = in_sizes[0] / 3;
    const int Nb = in_sizes[1] / 3;
    const int Nmax = Na > Nb ? Na : Nb;

    // constants mirroring the reference (NPOINTS=1024 is a module constant)
    const double rmaxd = 2.0 * sqrt(1.0 / (2.0 * sqrt(3.0) * 1024.0));
    const float rmax = (float)rmaxd;
    const float invR = (float)(1.0 / rmaxd);
    const float gf   = (float)(1.0 / (sqrt(M_PI) * 0.25));

    float* DmatT = (float*)d_ws;                          // [NbE][Na], sized Nmax*Na
    float* part  = DmatT + (size_t)Na * (size_t)Nmax;     // [JCH][NBBINS][Na]
    float* out   = (float*)d_out;

    // Kernel A: one 16x16 tile per wave
    const int jTmax = (Nmax + 15) / 16;
    const int iT    = (Na + 15) / 16;
    const int waves = jTmax * iT;
    const int blocksA = (waves * 32 + 127) / 128;
    pcf_dist_wmma<<<blocksA, 128, 0, stream>>>(dA, dB, same, DmatT, Na, Nb, invR, jTmax);

    // Kernel B: lane<->i, JCH deterministic j-chunks
    const int iBlocks = (Na + 255) / 256;
    pcf_hist<<<iBlocks * JCH, 256, 0, stream>>>(dA, dB, same, DmatT, part, Na, Nb, invR);

    // Kernel C: one block per radial bin
    pcf_finish<<<NBBINS, 256, 0, stream>>>(dA, part, out, Na, Nb, rmax, gf);
}